// GAT_25383256719662
// MI455X (gfx1250) — compile-verified
//
#include <hip/hip_runtime.h>

// Problem dimensions (fixed by the reference)
#define NNODES   20000
#define NEDGES   320000
#define DIN      78
#define HEADS    10
#define DOUT     128
#define NGRAPHS  128
#define ETOT     (NEDGES + NNODES)   // edges + self loops
#define NEGSLOPE 0.2f
#define EPS      1e-16f

typedef __attribute__((ext_vector_type(2))) float v2f;
typedef __attribute__((ext_vector_type(8))) float v8f;

// ---------- monotone float <-> uint order encoding (for atomicMax on floats) ---
__device__ __forceinline__ unsigned f2ord(float f) {
  unsigned u = __float_as_uint(f);
  return (u & 0x80000000u) ? ~u : (u | 0x80000000u);
}
__device__ __forceinline__ float ord2f(unsigned u) {
  return (u & 0x80000000u) ? __uint_as_float(u ^ 0x80000000u) : __uint_as_float(~u);
}
// f2ord(-FLT_MAX) = ~0xFF7FFFFF
#define NEGENC 0x00800000u

__device__ __forceinline__ float lrelu(float v) {
  return v > 0.0f ? v : NEGSLOPE * v;
}

__device__ __forceinline__ v8f wmma4(v2f a, v2f b, v8f c) {
  return __builtin_amdgcn_wmma_f32_16x16x4_f32(false, a, false, b, (short)0, c,
                                               false, false);
}

// ---------------- generic fill ------------------------------------------------
__global__ void fill_u32(unsigned* __restrict__ p, unsigned v, int n) {
  int t = blockIdx.x * blockDim.x + threadIdx.x;
  if (t < n) p[t] = v;
}

// ---------------- WMMA f32 GEMM: C[M,N] = A[M,K] @ B[K,N] (+bias)(+relu) ------
// One wave32 computes a 32x32 output macro-tile as 2x2 WMMA 16x16 tiles
// (V_WMMA_F32_16X16X4_F32, K stepped by 4).
// Out-of-range rows/cols are handled by ADDRESS CLAMPING: a duplicated A row
// only pollutes D rows >= M (never stored); a duplicated B col only pollutes
// D cols >= N (never stored). So the main K loop has NO predication and EXEC
// stays all-ones (WMMA requirement). Only the K tail (K % 4 != 0) is masked.
// Requires K even (all call sites: 78, 780, 128) so the per-lane A pair
// {k, k+1} is an aligned 8-byte load.
__global__ __launch_bounds__(256) void wmma_gemm_f32(
    const float* __restrict__ A, const float* __restrict__ B,
    float* __restrict__ C, const float* __restrict__ bias,
    int M, int N, int K, int relu) {
  const int lane = threadIdx.x & 31;
  const int wave = threadIdx.x >> 5;
  const int tilesN = (N + 31) >> 5;
  const int tilesM = (M + 31) >> 5;
  const int tile = blockIdx.x * (blockDim.x >> 5) + wave;
  if (tile >= tilesM * tilesN) return;      // wave-uniform exit
  const int tm = (tile / tilesN) << 5;      // macro-tile origin (rows)
  const int tn = (tile % tilesN) << 5;      // macro-tile origin (cols)
  const int half = lane >> 4;               // 0 or 1 (selects K pair)
  const int l16  = lane & 15;

  const int r0 = tm + l16, r1 = r0 + 16;
  const int c0 = tn + l16, c1 = c0 + 16;
  // clamped base pointers (see comment above: clamping is safe)
  const float* A0 = A + (size_t)(r0 < M ? r0 : M - 1) * K;
  const float* A1 = A + (size_t)(r1 < M ? r1 : M - 1) * K;
  const float* B0 = B + (c0 < N ? c0 : N - 1);
  const float* B1 = B + (c1 < N ? c1 : N - 1);

  v8f acc00 = {}, acc01 = {}, acc10 = {}, acc11 = {};
  const int kmain = K & ~3;
  for (int k0 = 0; k0 < kmain; k0 += 4) {
    const int k = k0 + half * 2;
    const v2f a0 = *(const v2f*)(A0 + k);          // global_load_b64
    const v2f a1 = *(const v2f*)(A1 + k);
    v2f b0, b1;
    b0.x = B0[(size_t)k * N];
    b0.y = B0[(size_t)(k + 1) * N];
    b1.x = B1[(size_t)k * N];
    b1.y = B1[(size_t)(k + 1) * N];
    acc00 = wmma4(a0, b0, acc00);
    acc01 = wmma4(a0, b1, acc01);
    acc10 = wmma4(a1, b0, acc10);
    acc11 = wmma4(a1, b1, acc11);
  }
  if (kmain < K) {                                 // masked K tail (once)
    const int k = kmain + half * 2;
    const bool k0ok = (k < K), k1ok = (k + 1 < K);
    const int ka = k0ok ? k : 0, kb = k1ok ? k + 1 : 0;
    v2f a0, a1, b0, b1;
    a0.x = k0ok ? A0[ka] : 0.0f;  a0.y = k1ok ? A0[kb] : 0.0f;
    a1.x = k0ok ? A1[ka] : 0.0f;  a1.y = k1ok ? A1[kb] : 0.0f;
    b0.x = k0ok ? B0[(size_t)ka * N] : 0.0f;
    b0.y = k1ok ? B0[(size_t)kb * N] : 0.0f;
    b1.x = k0ok ? B1[(size_t)ka * N] : 0.0f;
    b1.y = k1ok ? B1[(size_t)kb * N] : 0.0f;
    acc00 = wmma4(a0, b0, acc00);
    acc01 = wmma4(a0, b1, acc01);
    acc10 = wmma4(a1, b0, acc10);
    acc11 = wmma4(a1, b1, acc11);
  }

  // store 4 tiles (bounds-checked epilogue)
  {
    const v8f* accs[4] = {&acc00, &acc01, &acc10, &acc11};
    const int mb[4] = {tm, tm, tm + 16, tm + 16};
    const int cc[4] = {c0, c1, c0, c1};
#pragma unroll
    for (int t = 0; t < 4; ++t) {
      const int col = cc[t];
      if (col >= N) continue;
      const float bv = bias ? bias[col] : 0.0f;
#pragma unroll
      for (int i = 0; i < 8; ++i) {
        const int mrow = mb[t] + i + half * 8;
        if (mrow < M) {
          float v = (*accs[t])[i] + bv;
          if (relu) v = fmaxf(v, 0.0f);
          C[(size_t)mrow * N + col] = v;
        }
      }
    }
  }
}

// ---------------- per-(node,head) attention dots ------------------------------
__global__ void attn_dots(const float* __restrict__ h,
                          const float* __restrict__ att_src,
                          const float* __restrict__ att_dst,
                          float* __restrict__ a_src, float* __restrict__ a_dst,
                          int n_nodes, int H, int C) {
  int t = blockIdx.x * blockDim.x + threadIdx.x;
  if (t >= n_nodes * H) return;
  const int n = t / H, hh = t % H;
  const float* hp = h + ((size_t)n * H + hh) * C;
  const float* as = att_src + (size_t)hh * C;
  const float* ad = att_dst + (size_t)hh * C;
  float s0 = 0.0f, s1 = 0.0f;
  for (int c = 0; c < C; ++c) {
    float v = hp[c];
    s0 = fmaf(v, as[c], s0);
    s1 = fmaf(v, ad[c], s1);
  }
  a_src[t] = s0;
  a_dst[t] = s1;
}

// ---------------- softmax pass 1: segment max over dst ------------------------
__global__ void edge_max(const int* __restrict__ ei, int E,
                         const float* __restrict__ a_src,
                         const float* __restrict__ a_dst,
                         unsigned* __restrict__ m, int H, int etot) {
  int t = blockIdx.x * blockDim.x + threadIdx.x;
  if (t >= etot * H) return;
  const int e = t / H, hh = t % H;
  const int s = (e < E) ? ei[e] : (e - E);
  const int d = (e < E) ? ei[E + e] : (e - E);
  const float v = lrelu(a_src[s * H + hh] + a_dst[d * H + hh]);
  atomicMax(&m[d * H + hh], f2ord(v));
}

// ---------------- softmax pass 2: segment sum of exp --------------------------
__global__ void edge_sum(const int* __restrict__ ei, int E,
                         const float* __restrict__ a_src,
                         const float* __restrict__ a_dst,
                         const unsigned* __restrict__ m,
                         float* __restrict__ ssum, int H, int etot) {
  int t = blockIdx.x * blockDim.x + threadIdx.x;
  if (t >= etot * H) return;
  const int e = t / H, hh = t % H;
  const int s = (e < E) ? ei[e] : (e - E);
  const int d = (e < E) ? ei[E + e] : (e - E);
  const int idx = d * H + hh;
  const float v = lrelu(a_src[s * H + hh] + a_dst[idx]);
  atomicAdd(&ssum[idx], __expf(v - ord2f(m[idx])));
}

// ---------------- pass 3: out[dst] += alpha * h[src] (wave per edge-head) -----
__global__ __launch_bounds__(256) void edge_aggregate(
    const int* __restrict__ ei, int E,
    const float* __restrict__ a_src, const float* __restrict__ a_dst,
    const unsigned* __restrict__ m, const float* __restrict__ ssum,
    const float* __restrict__ h, float* __restrict__ out,
    int H, int C, int etot) {
  const int wave = (blockIdx.x * blockDim.x + threadIdx.x) >> 5;
  const int lane = threadIdx.x & 31;
  if (wave >= etot * H) return;
  const int e = wave / H, hh = wave % H;
  const int s = (e < E) ? ei[e] : (e - E);
  const int d = (e < E) ? ei[E + e] : (e - E);
  const int idx = d * H + hh;
  const float v = lrelu(a_src[s * H + hh] + a_dst[idx]);
  const float alpha = __expf(v - ord2f(m[idx])) / (ssum[idx] + EPS);
  const float* hs = h + ((size_t)s * H + hh) * C;
  float* od = out + ((size_t)d * H + hh) * C;
  for (int c = lane; c < C; c += 32)
    atomicAdd(&od[c], alpha * hs[c]);
}

// ---------------- bias + relu in place ---------------------------------------
__global__ void bias_relu(float* __restrict__ x, const float* __restrict__ b,
                          int n, int F) {
  int t = blockIdx.x * blockDim.x + threadIdx.x;
  if (t >= n * F) return;
  x[t] = fmaxf(x[t] + b[t % F], 0.0f);
}

// ---------------- global max-pool over sorted batch ---------------------------
__global__ void pool_max(const float* __restrict__ h, const int* __restrict__ batch,
                         unsigned* __restrict__ pooled, int n_nodes, int C) {
  int t = blockIdx.x * blockDim.x + threadIdx.x;
  if (t >= n_nodes * C) return;
  const int n = t / C, c = t % C;
  atomicMax(&pooled[batch[n] * C + c], f2ord(h[t]));
}

__global__ void decode_ord(const unsigned* __restrict__ u, float* __restrict__ f,
                           int n) {
  int t = blockIdx.x * blockDim.x + threadIdx.x;
  if (t < n) f[t] = ord2f(u[t]);
}

// =============================================================================
extern "C" void kernel_launch(void* const* d_in, const int* in_sizes, int n_in,
                              void* d_out, int out_size, void* d_ws, size_t ws_size,
                              hipStream_t stream) {
  (void)in_sizes; (void)n_in; (void)out_size; (void)ws_size;
  const float* x        = (const float*)d_in[0];
  const int*   ei       = (const int*)  d_in[1];  // [2, NEDGES] row-major
  const int*   batch    = (const int*)  d_in[2];
  const float* W1       = (const float*)d_in[3];
  const float* att_src1 = (const float*)d_in[4];
  const float* att_dst1 = (const float*)d_in[5];
  const float* b1       = (const float*)d_in[6];
  const float* W2       = (const float*)d_in[7];
  const float* att_src2 = (const float*)d_in[8];
  const float* att_dst2 = (const float*)d_in[9];
  const float* b2       = (const float*)d_in[10];
  const float* Wfc      = (const float*)d_in[11];
  const float* bfc      = (const float*)d_in[12];
  float* out = (float*)d_out;

  // --- carve workspace (256B aligned slabs) ---
  char* ws = (char*)d_ws;
  size_t off = 0;
  auto carve = [&](size_t elems) -> void* {
    void* p = ws + off;
    off += ((elems * 4) + 255) & ~(size_t)255;
    return p;
  };
  float*    h1   = (float*)   carve((size_t)NNODES * HEADS * DIN);  // 62.4 MB
  float*    agg1 = (float*)   carve((size_t)NNODES * HEADS * DIN);  // 62.4 MB
  float*    as1  = (float*)   carve((size_t)NNODES * HEADS);
  float*    ad1  = (float*)   carve((size_t)NNODES * HEADS);
  unsigned* m1   = (unsigned*)carve((size_t)NNODES * HEADS);
  float*    s1   = (float*)   carve((size_t)NNODES * HEADS);
  float*    h2   = (float*)   carve((size_t)NNODES * DOUT);         // 10.2 MB
  float*    agg2 = (float*)   carve((size_t)NNODES * DOUT);         // 10.2 MB
  float*    as2  = (float*)   carve((size_t)NNODES);
  float*    ad2  = (float*)   carve((size_t)NNODES);
  unsigned* m2   = (unsigned*)carve((size_t)NNODES);
  float*    s2   = (float*)   carve((size_t)NNODES);
  unsigned* pu   = (unsigned*)carve((size_t)NGRAPHS * DOUT);
  float*    pf   = (float*)   carve((size_t)NGRAPHS * DOUT);

  auto nblk = [](long long n, int t) { return (unsigned)((n + t - 1) / t); };

  // --- init accumulators / reduction state (every call: deterministic) ---
  fill_u32<<<nblk((long long)NNODES*HEADS*DIN,256),256,0,stream>>>((unsigned*)agg1, 0u, NNODES*HEADS*DIN);
  fill_u32<<<nblk((long long)NNODES*DOUT,256),256,0,stream>>>((unsigned*)agg2, 0u, NNODES*DOUT);
  fill_u32<<<nblk(NNODES*HEADS,256),256,0,stream>>>((unsigned*)s1, 0u, NNODES*HEADS);
  fill_u32<<<nblk(NNODES,256),256,0,stream>>>((unsigned*)s2, 0u, NNODES);
  fill_u32<<<nblk(NNODES*HEADS,256),256,0,stream>>>(m1, NEGENC, NNODES*HEADS);
  fill_u32<<<nblk(NNODES,256),256,0,stream>>>(m2, NEGENC, NNODES);
  fill_u32<<<nblk(NGRAPHS*DOUT,256),256,0,stream>>>(pu, NEGENC, NGRAPHS*DOUT);

  // ================= layer 1: GATConv(78 -> 10 x 78) =================
  {
    const int M = NNODES, N = HEADS * DIN, K = DIN;
    const int tiles = ((M + 31) / 32) * ((N + 31) / 32);
    wmma_gemm_f32<<<nblk(tiles, 8), 256, 0, stream>>>(x, W1, h1, nullptr, M, N, K, 0);
  }
  attn_dots<<<nblk(NNODES*HEADS,256),256,0,stream>>>(h1, att_src1, att_dst1, as1, ad1,
                                                     NNODES, HEADS, DIN);
  edge_max<<<nblk((long long)ETOT*HEADS,256),256,0,stream>>>(ei, NEDGES, as1, ad1, m1, HEADS, ETOT);
  edge_sum<<<nblk((long long)ETOT*HEADS,256),256,0,stream>>>(ei, NEDGES, as1, ad1, m1, s1, HEADS, ETOT);
  edge_aggregate<<<nblk((long long)ETOT*HEADS*32,256),256,0,stream>>>(
      ei, NEDGES, as1, ad1, m1, s1, h1, agg1, HEADS, DIN, ETOT);
  bias_relu<<<nblk((long long)NNODES*HEADS*DIN,256),256,0,stream>>>(agg1, b1, NNODES, HEADS*DIN);

  // ================= layer 2: GATConv(780 -> 1 x 128) =================
  {
    const int M = NNODES, N = DOUT, K = HEADS * DIN;
    const int tiles = ((M + 31) / 32) * ((N + 31) / 32);
    wmma_gemm_f32<<<nblk(tiles, 8), 256, 0, stream>>>(agg1, W2, h2, nullptr, M, N, K, 0);
  }
  attn_dots<<<nblk(NNODES,256),256,0,stream>>>(h2, att_src2, att_dst2, as2, ad2,
                                               NNODES, 1, DOUT);
  edge_max<<<nblk(ETOT,256),256,0,stream>>>(ei, NEDGES, as2, ad2, m2, 1, ETOT);
  edge_sum<<<nblk(ETOT,256),256,0,stream>>>(ei, NEDGES, as2, ad2, m2, s2, 1, ETOT);
  edge_aggregate<<<nblk((long long)ETOT*32,256),256,0,stream>>>(
      ei, NEDGES, as2, ad2, m2, s2, h2, agg2, 1, DOUT, ETOT);
  bias_relu<<<nblk((long long)NNODES*DOUT,256),256,0,stream>>>(agg2, b2, NNODES, DOUT);

  // ================= global max pool + FC =================
  pool_max<<<nblk((long long)NNODES*DOUT,256),256,0,stream>>>(agg2, batch, pu, NNODES, DOUT);
  decode_ord<<<nblk(NGRAPHS*DOUT,256),256,0,stream>>>(pu, pf, NGRAPHS*DOUT);
  {
    const int M = NGRAPHS, N = DOUT, K = DOUT;
    const int tiles = ((M + 31) / 32) * ((N + 31) / 32);   // 16 macro-tiles
    wmma_gemm_f32<<<nblk(tiles, 8), 256, 0, stream>>>(pf, Wfc, out, bfc, M, N, K, 1);
  }
}